// QWRFNet_54924041781640
// MI455X (gfx1250) — compile-verified
//
#include <hip/hip_runtime.h>
#include <math.h>

#define B_ 2
#define CH_ 64
#define H0_ 256
#define W0_ 256
#define HW0_ (H0_ * W0_)
#define TPB 256

typedef __attribute__((ext_vector_type(2))) float v2f;
typedef __attribute__((ext_vector_type(8))) float v8f;

static inline int cdivll(long long a, long long b) { return (int)((a + b - 1) / b); }

// ---------------------------------------------------------------------------
// Channel LayerNorm (normalize over C at each pixel)
// ---------------------------------------------------------------------------
__global__ __launch_bounds__(TPB) void k_cln(const float* __restrict__ x,
                                             const float* __restrict__ g,
                                             const float* __restrict__ bt,
                                             float* __restrict__ out,
                                             int C, int HW, long long total) {
  long long idx = (long long)blockIdx.x * blockDim.x + threadIdx.x;
  if (idx >= total) return;
  int b = (int)(idx / HW);
  int s = (int)(idx % HW);
  const float* xp = x + (size_t)b * C * HW + s;
  float mean = 0.f;
  for (int c = 0; c < C; ++c) mean += xp[(size_t)c * HW];
  mean /= (float)C;
  float var = 0.f;
  for (int c = 0; c < C; ++c) { float d = xp[(size_t)c * HW] - mean; var += d * d; }
  var /= (float)C;
  float inv = rsqrtf(var + 1e-6f);
  float* op = out + (size_t)b * C * HW + s;
  for (int c = 0; c < C; ++c)
    op[(size_t)c * HW] = (xp[(size_t)c * HW] - mean) * inv * g[c] + bt[c];
}

// ---------------------------------------------------------------------------
// add_coords + zero-pad to window multiple: out (B, C+2, Hp, Wp)
// ---------------------------------------------------------------------------
__global__ __launch_bounds__(TPB) void k_addcoords_pad(const float* __restrict__ x,
                                                       float* __restrict__ out,
                                                       int C, int H, int W, int Hp, int Wp,
                                                       long long total) {
  long long idx = (long long)blockIdx.x * blockDim.x + threadIdx.x;
  if (idx >= total) return;
  int Cc = C + 2;
  int xw = (int)(idx % Wp);
  long long r = idx / Wp;
  int y = (int)(r % Hp); r /= Hp;
  int c = (int)(r % Cc);
  int b = (int)(r / Cc);
  float v = 0.f;
  if (y < H && xw < W) {
    if (c < C)       v = x[(((size_t)b * C + c) * H + y) * W + xw];
    else if (c == C) v = -1.f + 2.f * (float)xw / (float)(W - 1);
    else             v = -1.f + 2.f * (float)y  / (float)(H - 1);
  }
  out[idx] = v;
}

// ---------------------------------------------------------------------------
// Generic 1x1-conv GEMM on WMMA f32 16x16x4, 2x2 register tiling per wave.
//   out[b, m, s] = act( sum_k W[m,k] * in(b,k,s) + bias[m] ) + resid[b,m,s]
// Input channels may be a concat of up to 3 tensors (c0+c1+c2 = Cin), processed
// as K segments so the hot loop has a uniform base pointer and zero masking.
// Each wave computes a 32(M) x 32(N) output tile: 4 accumulators, 4 WMMAs and
// 8 loads per K-step of 4.
// ---------------------------------------------------------------------------
__global__ __launch_bounds__(TPB) void k_gemm(const float* __restrict__ in0, int c0,
                                              const float* __restrict__ in1, int c1,
                                              const float* __restrict__ in2, int c2,
                                              const float* __restrict__ Wt,
                                              const float* __restrict__ bias,
                                              const float* __restrict__ resid,
                                              float* __restrict__ out,
                                              int Cout, int HW, int Bn, int act,
                                              int tilesM, long long tilesTotal) {
  long long wave = (long long)blockIdx.x * (blockDim.x >> 5) + (threadIdx.x >> 5);
  if (wave >= tilesTotal) return;  // wave-uniform exit: surviving waves keep full EXEC
  int lane = threadIdx.x & 31;
  int lm   = lane & 15;
  int half = lane >> 4;              // K half: 0 -> K{0,1}, 1 -> K{2,3}
  int tm = (int)(wave % tilesM);
  long long tn = wave / tilesM;
  int Cin = c0 + c1 + c2;
  long long N = (long long)Bn * HW;

  // Two M rows (16 apart) per lane, clamped (garbage rows masked at store).
  int m0 = tm * 32 + lm;
  int m1 = m0 + 16;
  int m0c = m0 < Cout ? m0 : (Cout - 1);
  int m1c = m1 < Cout ? m1 : (Cout - 1);

  // Two N column groups (16 apart), clamped (garbage cols masked at store).
  long long col0 = tn * 32 + lm;
  long long col1 = col0 + 16;
  bool vn0 = col0 < N, vn1 = col1 < N;
  long long cc0 = vn0 ? col0 : (N - 1);
  long long cc1 = vn1 ? col1 : (N - 1);
  int b0 = (int)(cc0 / HW), s0 = (int)(cc0 % HW);
  int b1 = (int)(cc1 / HW), s1 = (int)(cc1 % HW);

  v8f a00, a01, a10, a11;  // acc[m-tile][n-tile]
#pragma unroll
  for (int i = 0; i < 8; ++i) { a00[i] = 0.f; a01[i] = 0.f; a10[i] = 0.f; a11[i] = 0.f; }

  const float* segp[3] = {in0, in1, in2};
  int segc[3] = {c0, c1, c2};
  int segoff = 0;
  size_t sHW = (size_t)HW;

  for (int sg = 0; sg < 3; ++sg) {
    int Cs = segc[sg];
    if (Cs <= 0) continue;
    const float* P = segp[sg];
    // per-lane streaming pointers, pre-offset by this lane's K-half
    const float* w0  = Wt + (size_t)m0c * Cin + segoff + 2 * half;
    const float* w1  = Wt + (size_t)m1c * Cin + segoff + 2 * half;
    const float* p0  = P + ((size_t)b0 * Cs + 2 * half) * sHW + s0;
    const float* p1  = P + ((size_t)b1 * Cs + 2 * half) * sHW + s1;
    int kmain = Cs & ~3;
#pragma unroll 2
    for (int k0 = 0; k0 < kmain; k0 += 4) {
      __builtin_prefetch(p0 + 16 * sHW, 0, 3);
      v2f af0, af1, bf0, bf1;
      af0[0] = w0[0];   af0[1] = w0[1];
      af1[0] = w1[0];   af1[1] = w1[1];
      bf0[0] = p0[0];   bf0[1] = p0[sHW];
      bf1[0] = p1[0];   bf1[1] = p1[sHW];
      a00 = __builtin_amdgcn_wmma_f32_16x16x4_f32(false, af0, false, bf0, (short)0, a00, false, false);
      a01 = __builtin_amdgcn_wmma_f32_16x16x4_f32(false, af0, false, bf1, (short)0, a01, false, false);
      a10 = __builtin_amdgcn_wmma_f32_16x16x4_f32(false, af1, false, bf0, (short)0, a10, false, false);
      a11 = __builtin_amdgcn_wmma_f32_16x16x4_f32(false, af1, false, bf1, (short)0, a11, false, false);
      w0 += 4; w1 += 4; p0 += 4 * sHW; p1 += 4 * sHW;
    }
    if (kmain < Cs) {  // K tail (Cs % 4 != 0): masked, runs once per segment
      int ka = kmain + 2 * half;
      int kb = ka + 1;
      float fa = ka < Cs ? 1.f : 0.f;
      float fb = kb < Cs ? 1.f : 0.f;
      int kac = ka < Cs ? ka : (Cs - 1);
      int kbc = kb < Cs ? kb : (Cs - 1);
      const float* w0b = Wt + (size_t)m0c * Cin + segoff;
      const float* w1b = Wt + (size_t)m1c * Cin + segoff;
      const float* p0b = P + (size_t)b0 * Cs * sHW + s0;
      const float* p1b = P + (size_t)b1 * Cs * sHW + s1;
      v2f af0, af1, bf0, bf1;
      af0[0] = w0b[kac] * fa;        af0[1] = w0b[kbc] * fb;
      af1[0] = w1b[kac] * fa;        af1[1] = w1b[kbc] * fb;
      bf0[0] = p0b[(size_t)kac * sHW] * fa;  bf0[1] = p0b[(size_t)kbc * sHW] * fb;
      bf1[0] = p1b[(size_t)kac * sHW] * fa;  bf1[1] = p1b[(size_t)kbc * sHW] * fb;
      a00 = __builtin_amdgcn_wmma_f32_16x16x4_f32(false, af0, false, bf0, (short)0, a00, false, false);
      a01 = __builtin_amdgcn_wmma_f32_16x16x4_f32(false, af0, false, bf1, (short)0, a01, false, false);
      a10 = __builtin_amdgcn_wmma_f32_16x16x4_f32(false, af1, false, bf0, (short)0, a10, false, false);
      a11 = __builtin_amdgcn_wmma_f32_16x16x4_f32(false, af1, false, bf1, (short)0, a11, false, false);
    }
    segoff += Cs;
  }

  // Store with bias / activation / residual fusion.
#pragma unroll
  for (int r = 0; r < 8; ++r) {
    int mm0 = tm * 32 + r + 8 * half;  // D layout: VGPR r -> M = r (+8 upper half-wave)
    int mm1 = mm0 + 16;
    float vals[4] = {a00[r], a01[r], a10[r], a11[r]};
    int   ms[4]   = {mm0, mm0, mm1, mm1};
    bool  vns[4]  = {vn0, vn1, vn0, vn1};
    int   bs[4]   = {b0, b1, b0, b1};
    int   ss[4]   = {s0, s1, s0, s1};
#pragma unroll
    for (int q = 0; q < 4; ++q) {
      int m = ms[q];
      if (vns[q] && m < Cout) {
        float v = vals[q];
        if (bias) v += bias[m];
        if (act == 1)      v = 0.5f * v * (1.f + erff(v * 0.70710678118654752f));  // exact gelu
        else if (act == 2) v = 1.f / (1.f + __expf(-v));                           // sigmoid
        size_t o = ((size_t)bs[q] * Cout + m) * sHW + ss[q];
        if (resid) v += resid[o];
        out[o] = v;
      }
    }
  }
}

// ---------------------------------------------------------------------------
// Window attention: one block per 7x7 window. qkv: (B, 3*Cc, Hp, Wp),
// out: (B, Cc, H, W) (cropped). LDS: q[49*Cc] (reused for v), k[49*Cc], attn[49*49].
// ---------------------------------------------------------------------------
extern __shared__ float smem[];
__global__ __launch_bounds__(TPB) void k_winattn(const float* __restrict__ qkv,
                                                 float* __restrict__ out,
                                                 int Cc, int Hp, int Wp, int H, int W,
                                                 int nH, int nW, float scale) {
  const int WS2 = 49;
  float* qs = smem;
  float* ks = qs + WS2 * Cc;
  float* at = ks + WS2 * Cc;
  int win = blockIdx.x;
  int b = win / (nH * nW);
  int r = win % (nH * nW);
  int wy = (r / nW) * 7, wx = (r % nW) * 7;
  size_t plane = (size_t)Hp * Wp;
  const float* qb = qkv + (size_t)b * 3 * Cc * plane;
  int tid = threadIdx.x;

  for (int t = tid; t < WS2 * Cc; t += blockDim.x) {
    int j = t % WS2, c = t / WS2;
    int y = wy + j / 7, x = wx + j % 7;
    size_t off = (size_t)c * plane + (size_t)y * Wp + x;
    qs[j * Cc + c] = qb[off];
    ks[j * Cc + c] = qb[off + (size_t)Cc * plane];
  }
  __syncthreads();
  for (int t = tid; t < WS2 * WS2; t += blockDim.x) {
    int i = t / WS2, j = t % WS2;
    const float* qi = qs + i * Cc;
    const float* kj = ks + j * Cc;
    float d = 0.f;
    for (int c = 0; c < Cc; ++c) d += qi[c] * kj[c];
    at[t] = d * scale;
  }
  __syncthreads();
  if (tid < WS2) {
    float* row = at + tid * WS2;
    float mx = row[0];
    for (int j = 1; j < WS2; ++j) mx = fmaxf(mx, row[j]);
    float sum = 0.f;
    for (int j = 0; j < WS2; ++j) { float e = __expf(row[j] - mx); row[j] = e; sum += e; }
    float inv = 1.f / sum;
    for (int j = 0; j < WS2; ++j) row[j] *= inv;
  }
  __syncthreads();
  // reload V over qs
  for (int t = tid; t < WS2 * Cc; t += blockDim.x) {
    int j = t % WS2, c = t / WS2;
    int y = wy + j / 7, x = wx + j % 7;
    qs[j * Cc + c] = qb[(size_t)(2 * Cc + c) * plane + (size_t)y * Wp + x];
  }
  __syncthreads();
  for (int t = tid; t < WS2 * Cc; t += blockDim.x) {
    int i = t % WS2, c = t / WS2;
    int y = wy + i / 7, x = wx + i % 7;
    if (y < H && x < W) {
      const float* ar = at + i * WS2;
      float acc = 0.f;
      for (int j = 0; j < WS2; ++j) acc += qs[j * Cc + c] * ar[j];
      out[((size_t)b * Cc + c) * ((size_t)H * W) + (size_t)y * W + x] = acc;
    }
  }
}

// ---------------------------------------------------------------------------
// Dense 3x3 conv, pad=1, arbitrary stride (dm_conv / ds_conv downsamplers)
// ---------------------------------------------------------------------------
__global__ __launch_bounds__(TPB) void k_conv3x3(const float* __restrict__ in,
                                                 const float* __restrict__ w,
                                                 const float* __restrict__ bias,
                                                 float* __restrict__ out,
                                                 int Cin, int Cout, int H, int W,
                                                 int Ho, int Wo, int stride, long long total) {
  long long idx = (long long)blockIdx.x * blockDim.x + threadIdx.x;
  if (idx >= total) return;
  int ox = (int)(idx % Wo); long long r = idx / Wo;
  int oy = (int)(r % Ho); r /= Ho;
  int o = (int)(r % Cout);
  int b = (int)(r / Cout);
  float acc = bias ? bias[o] : 0.f;
  int iy0 = oy * stride - 1, ix0 = ox * stride - 1;
  const float* wb = w + (size_t)o * Cin * 9;
  for (int ci = 0; ci < Cin; ++ci) {
    const float* ip = in + ((size_t)b * Cin + ci) * H * W;
    const float* wc = wb + ci * 9;
    for (int ky = 0; ky < 3; ++ky) {
      int y = iy0 + ky; if (y < 0 || y >= H) continue;
      for (int kx = 0; kx < 3; ++kx) {
        int x = ix0 + kx; if (x < 0 || x >= W) continue;
        acc += ip[(size_t)y * W + x] * wc[ky * 3 + kx];
      }
    }
  }
  out[idx] = acc;
}

// ---------------------------------------------------------------------------
// Grouped depthwise 3x3 (stride 1, pad 1), channel multiplier m (Cout=Cin*m)
// ---------------------------------------------------------------------------
__global__ __launch_bounds__(TPB) void k_dwconv(const float* __restrict__ in,
                                                const float* __restrict__ w,
                                                const float* __restrict__ bias,
                                                float* __restrict__ out,
                                                int Cin, int mult, int H, int W,
                                                long long total) {
  long long idx = (long long)blockIdx.x * blockDim.x + threadIdx.x;
  if (idx >= total) return;
  int Cout = Cin * mult;
  int x = (int)(idx % W); long long r = idx / W;
  int y = (int)(r % H); r /= H;
  int o = (int)(r % Cout);
  int b = (int)(r / Cout);
  int ci = o / mult;
  float acc = bias ? bias[o] : 0.f;
  const float* ip = in + ((size_t)b * Cin + ci) * H * W;
  const float* wc = w + (size_t)o * 9;
  for (int ky = 0; ky < 3; ++ky) {
    int yy = y + ky - 1; if (yy < 0 || yy >= H) continue;
    for (int kx = 0; kx < 3; ++kx) {
      int xx = x + kx - 1; if (xx < 0 || xx >= W) continue;
      acc += ip[(size_t)yy * W + xx] * wc[ky * 3 + kx];
    }
  }
  out[idx] = acc;
}

// ---------------------------------------------------------------------------
// Bilinear resize, half-pixel centers (align_corners=False upsample)
// ---------------------------------------------------------------------------
__global__ __launch_bounds__(TPB) void k_resize(const float* __restrict__ in,
                                                float* __restrict__ out, int C,
                                                int Hin, int Win, int Hout, int Wout,
                                                long long total) {
  long long idx = (long long)blockIdx.x * blockDim.x + threadIdx.x;
  if (idx >= total) return;
  int ox = (int)(idx % Wout); long long r = idx / Wout;
  int oy = (int)(r % Hout); r /= Hout;
  int c = (int)(r % C);
  int b = (int)(r / C);
  float sy = ((float)oy + 0.5f) * (float)Hin / (float)Hout - 0.5f;
  float sx = ((float)ox + 0.5f) * (float)Win / (float)Wout - 0.5f;
  int y0 = (int)floorf(sy); float wy = sy - (float)y0;
  int x0 = (int)floorf(sx); float wx = sx - (float)x0;
  int y1 = y0 + 1, x1 = x0 + 1;
  y0 = y0 < 0 ? 0 : (y0 >= Hin ? Hin - 1 : y0);
  y1 = y1 < 0 ? 0 : (y1 >= Hin ? Hin - 1 : y1);
  x0 = x0 < 0 ? 0 : (x0 >= Win ? Win - 1 : x0);
  x1 = x1 < 0 ? 0 : (x1 >= Win ? Win - 1 : x1);
  const float* ip = in + ((size_t)b * C + c) * Hin * Win;
  float v = (1.f - wy) * ((1.f - wx) * ip[(size_t)y0 * Win + x0] + wx * ip[(size_t)y0 * Win + x1]) +
            wy        * ((1.f - wx) * ip[(size_t)y1 * Win + x0] + wx * ip[(size_t)y1 * Win + x1]);
  out[idx] = v;
}

// ---------------------------------------------------------------------------
// Elementwise helpers
// ---------------------------------------------------------------------------
__global__ __launch_bounds__(TPB) void k_add(const float* a, const float* b, float* o, long long n) {
  long long i = (long long)blockIdx.x * blockDim.x + threadIdx.x;
  if (i < n) o[i] = a[i] + b[i];
}
__global__ __launch_bounds__(TPB) void k_mul(const float* a, const float* b, float* o, long long n) {
  long long i = (long long)blockIdx.x * blockDim.x + threadIdx.x;
  if (i < n) o[i] = a[i] * b[i];
}
// relu(a)*b where ab = [a(Ch) ; b(Ch)] channels (single batch image)
__global__ __launch_bounds__(TPB) void k_glu(const float* __restrict__ ab, float* __restrict__ g,
                                             int Ch, int HW, long long n) {
  long long i = (long long)blockIdx.x * blockDim.x + threadIdx.x;
  if (i >= n) return;
  int c = (int)(i / HW), s = (int)(i % HW);
  float a = ab[(size_t)c * HW + s];
  float b = ab[(size_t)(Ch + c) * HW + s];
  g[i] = (a > 0.f ? a : 0.f) * b;
}
// z = (1-a)*x + a*y, a is (B,1,H,W)
__global__ __launch_bounds__(TPB) void k_mix(const float* __restrict__ x, const float* __restrict__ y,
                                             const float* __restrict__ a, float* __restrict__ z,
                                             int C, int HW, long long n) {
  long long i = (long long)blockIdx.x * blockDim.x + threadIdx.x;
  if (i >= n) return;
  int b = (int)(i / ((long long)C * HW));
  int s = (int)(i % HW);
  float av = a[(size_t)b * HW + s];
  z[i] = (1.f - av) * x[i] + av * y[i];
}

// ===========================================================================
// Host side
// ===========================================================================
struct Ctx {
  hipStream_t st;
  char* base;
  size_t off;
  float* alloc(long long n) {
    float* p = (float*)(base + off);
    off += (((size_t)n * 4) + 255) & ~(size_t)255;
    return p;
  }
};

static void gemm(Ctx& cx, const float* in0, int c0, const float* in1, int c1,
                 const float* in2, int c2, const float* Wt, const float* bias,
                 const float* resid, float* out, int Cout, int HW, int Bn, int act) {
  if (!in1) { in1 = in0; c1 = 0; }
  if (!in2) { in2 = in0; c2 = 0; }
  long long N = (long long)Bn * HW;
  int tm = (Cout + 31) / 32;                 // 32 rows of M per wave (2 tiles)
  long long tn = (N + 31) / 32;              // 32 cols of N per wave (2 tiles)
  long long tiles = (long long)tm * tn;
  int blocks = cdivll(tiles, 8);
  k_gemm<<<blocks, TPB, 0, cx.st>>>(in0, c0, in1, c1, in2, c2, Wt, bias, resid, out,
                                    Cout, HW, Bn, act, tm, tiles);
}

static void cln(Ctx& cx, const float* x, const float* g, const float* b, float* out,
                int C, int HW, int Bn) {
  long long total = (long long)Bn * HW;
  k_cln<<<cdivll(total, TPB), TPB, 0, cx.st>>>(x, g, b, out, C, HW, total);
}

static void wattn(Ctx& cx, const float* x, int C, int H, int W, const float* qkv_w,
                  const float* proj_w, const float* proj_b, float* out) {
  int Cc = C + 2;
  int nH = (H + 6) / 7, nW = (W + 6) / 7;
  int Hp = nH * 7, Wp = nW * 7;
  size_t mk = cx.off;
  long long npad = (long long)B_ * Cc * Hp * Wp;
  float* xc = cx.alloc(npad);
  k_addcoords_pad<<<cdivll(npad, TPB), TPB, 0, cx.st>>>(x, xc, C, H, W, Hp, Wp, npad);
  float* qkv = cx.alloc((long long)B_ * 3 * Cc * Hp * Wp);
  gemm(cx, xc, Cc, 0, 0, 0, 0, qkv_w, nullptr, nullptr, qkv, 3 * Cc, Hp * Wp, B_, 0);
  float* att = cx.alloc((long long)B_ * Cc * H * W);
  int lds = (2 * 49 * Cc + 49 * 49) * 4;
  hipFuncSetAttribute((const void*)k_winattn, hipFuncAttributeMaxDynamicSharedMemorySize, lds);
  k_winattn<<<B_ * nH * nW, TPB, lds, cx.st>>>(qkv, att, Cc, Hp, Wp, H, W, nH, nW,
                                               1.0f / sqrtf((float)Cc));
  gemm(cx, att, Cc, 0, 0, 0, 0, proj_w, proj_b, nullptr, out, C, H * W, B_, 0);
  cx.off = mk;
}

struct MsmP {
  const float *lg_proj_b, *lg_proj_w, *lg_qkv_w;
  const float *md_proj_b, *md_proj_w, *md_qkv_w;
  const float *sm_proj_b, *sm_proj_w, *sm_qkv_w;
  const float *cmd_b, *cmd_w, *csm_b, *csm_w;
  const float *dmc_b, *dmc_w, *dmn_b, *dmn_w;
  const float *dsc_b, *dsc_w, *dsn_b, *dsn_w;
  const float *fuse_b, *fuse_w;
  const float *ffdw_b, *ffdw_w, *ffp1_b, *ffp1_w, *ffp2_b, *ffp2_w;
  const float *n1_b, *n1_w, *n2_b, *n2_w;
};

static MsmP load_msm(void* const* din, int base) {
  MsmP P;
  const float** q = (const float**)&P;
  for (int i = 0; i < 33; ++i) q[i] = (const float*)din[base + i];
  return P;
}

static void msatten(Ctx& cx, const float* x, const float* resid, const MsmP& P, float* out) {
  const int HW = HW0_;
  size_t mk = cx.off;
  float* LG = cx.alloc((long long)B_ * 64 * HW);
  wattn(cx, x, 64, 256, 256, P.lg_qkv_w, P.lg_proj_w, P.lg_proj_b, LG);

  float* MD = cx.alloc((long long)B_ * 64 * HW);
  {
    size_t mk1 = cx.off;
    float* tn = cx.alloc((long long)B_ * 64 * HW);
    cln(cx, x, P.dmn_w, P.dmn_b, tn, 64, HW, B_);
    long long no = (long long)B_ * 128 * 128 * 128;
    float* mdin = cx.alloc(no);
    k_conv3x3<<<cdivll(no, TPB), TPB, 0, cx.st>>>(tn, P.dmc_w, P.dmc_b, mdin,
                                                  64, 128, 256, 256, 128, 128, 2, no);
    float* wa = cx.alloc(no);
    wattn(cx, mdin, 128, 128, 128, P.md_qkv_w, P.md_proj_w, P.md_proj_b, wa);
    float* mdc = cx.alloc((long long)B_ * 64 * 128 * 128);
    gemm(cx, wa, 128, 0, 0, 0, 0, P.cmd_w, P.cmd_b, nullptr, mdc, 64, 128 * 128, B_, 0);
    long long nr = (long long)B_ * 64 * HW;
    k_resize<<<cdivll(nr, TPB), TPB, 0, cx.st>>>(mdc, MD, 64, 128, 128, 256, 256, nr);
    cx.off = mk1;
  }

  float* SM = cx.alloc((long long)B_ * 64 * HW);
  {
    size_t mk1 = cx.off;
    float* tn = cx.alloc((long long)B_ * 64 * HW);
    cln(cx, x, P.dsn_w, P.dsn_b, tn, 64, HW, B_);
    long long no = (long long)B_ * 256 * 64 * 64;
    float* smin = cx.alloc(no);
    k_conv3x3<<<cdivll(no, TPB), TPB, 0, cx.st>>>(tn, P.dsc_w, P.dsc_b, smin,
                                                  64, 256, 256, 256, 64, 64, 4, no);
    float* wa = cx.alloc(no);
    wattn(cx, smin, 256, 64, 64, P.sm_qkv_w, P.sm_proj_w, P.sm_proj_b, wa);
    float* smc = cx.alloc((long long)B_ * 64 * 64 * 64);
    gemm(cx, wa, 256, 0, 0, 0, 0, P.csm_w, P.csm_b, nullptr, smc, 64, 64 * 64, B_, 0);
    long long nr = (long long)B_ * 64 * HW;
    k_resize<<<cdivll(nr, TPB), TPB, 0, cx.st>>>(smc, SM, 64, 64, 64, 256, 256, nr);
    cx.off = mk1;
  }

  // fuse(concat[lg, md, sm]) + residual (pre-norm input)
  gemm(cx, LG, 64, MD, 64, SM, 64, P.fuse_w, P.fuse_b, resid, out, 64, HW, B_, 0);
  cx.off = mk;
}

extern "C" void kernel_launch(void* const* d_in, const int* in_sizes, int n_in,
                              void* d_out, int out_size, void* d_ws, size_t ws_size,
                              hipStream_t stream) {
  (void)in_sizes; (void)n_in; (void)out_size; (void)ws_size;
  Ctx cx{stream, (char*)d_ws, 0};
  const int HW = HW0_;
  const long long NT = (long long)B_ * 64 * HW;

  // Input map: [0]=x; params flattened in canonical (sorted-key) order.
  const float* x = (const float*)d_in[0];
  const float* adj_b[2] = {(const float*)d_in[1], (const float*)d_in[3]};
  const float* adj_w[2] = {(const float*)d_in[2], (const float*)d_in[4]};
  const float *aw_ca_b = (const float*)d_in[5],  *aw_ca_w = (const float*)d_in[6];
  const float *aw_cx_b = (const float*)d_in[7],  *aw_cx_w = (const float*)d_in[8];
  const float *aw_cy_b = (const float*)d_in[9],  *aw_cy_w = (const float*)d_in[10];
  const float *aw_fu_b = (const float*)d_in[11], *aw_fu_w = (const float*)d_in[12];
  const float *mc_dw1_b = (const float*)d_in[13], *mc_dw1_w = (const float*)d_in[14];
  const float *mc_dw2_b = (const float*)d_in[15], *mc_dw2_w = (const float*)d_in[16];
  const float *mc_nm_b  = (const float*)d_in[17], *mc_nm_w  = (const float*)d_in[18];
  const float *mc_pw1_b = (const float*)d_in[19], *mc_pw1_w = (const float*)d_in[20];
  const float *mc_pw2_b = (const float*)d_in[21], *mc_pw2_w = (const float*)d_in[22];

  float* curbuf[2] = {cx.alloc(NT), cx.alloc(NT)};
  const float* cur = x;

  // ---- MSM blocks + adjust convs ----
  for (int i = 0; i < 2; ++i) {
    MsmP P = load_msm(d_in, 23 + 33 * i);
    size_t mk = cx.off;
    float* t1 = cx.alloc(NT);
    cln(cx, cur, P.n1_w, P.n1_b, t1, 64, HW, B_);
    float* x1 = cx.alloc(NT);
    msatten(cx, t1, cur, P, x1);  // x1 = cur + msatten(cln(cur))
    float* t2 = cx.alloc(NT);
    cln(cx, x1, P.n2_w, P.n2_b, t2, 64, HW, B_);
    float* x2 = cx.alloc(NT);
    for (int b = 0; b < B_; ++b) {  // FFN, chunked per batch image to bound scratch
      size_t mk2 = cx.off;
      const float* t2b = t2 + (size_t)b * 64 * HW;
      float* h = cx.alloc((long long)256 * HW);
      gemm(cx, t2b, 64, 0, 0, 0, 0, P.ffp1_w, P.ffp1_b, nullptr, h, 256, HW, 1, 1);
      float* h2 = cx.alloc((long long)256 * HW);
      long long nd = (long long)256 * HW;
      k_dwconv<<<cdivll(nd, TPB), TPB, 0, stream>>>(h, P.ffdw_w, P.ffdw_b, h2,
                                                    256, 1, 256, 256, nd);
      gemm(cx, h2, 256, 0, 0, 0, 0, P.ffp2_w, P.ffp2_b, x1 + (size_t)b * 64 * HW,
           x2 + (size_t)b * 64 * HW, 64, HW, 1, 0);
      cx.off = mk2;
    }
    // adj: conv1x1(concat[x2, cur])
    gemm(cx, x2, 64, cur, 64, 0, 0, adj_w[i], adj_b[i], nullptr, curbuf[i], 64, HW, B_, 0);
    cur = curbuf[i];
    cx.off = mk;
  }

  // ---- MCM branch on original x ----
  size_t mk = cx.off;
  float* sc   = cx.alloc(NT);
  float* mout = cx.alloc(NT);
  cln(cx, x, mc_nm_w, mc_nm_b, sc, 64, HW, B_);
  for (int b = 0; b < B_; ++b) {
    size_t mk2 = cx.off;
    const float* scb = sc + (size_t)b * 64 * HW;
    long long n384 = (long long)384 * HW;
    float* dwt = cx.alloc(n384);
    k_dwconv<<<cdivll(n384, TPB), TPB, 0, stream>>>(scb, mc_dw1_w, mc_dw1_b, dwt,
                                                    64, 6, 256, 256, n384);
    float* ab = cx.alloc(n384);
    gemm(cx, dwt, 384, 0, 0, 0, 0, mc_pw1_w, mc_pw1_b, nullptr, ab, 384, HW, 1, 0);
    long long n192 = (long long)192 * HW;
    float* g = cx.alloc(n192);
    k_glu<<<cdivll(n192, TPB), TPB, 0, stream>>>(ab, g, 192, HW, n192);
    float* t2 = cx.alloc(n192);
    k_dwconv<<<cdivll(n192, TPB), TPB, 0, stream>>>(g, mc_dw2_w, mc_dw2_b, t2,
                                                    192, 1, 256, 256, n192);
    gemm(cx, t2, 192, 0, 0, 0, 0, mc_pw2_w, mc_pw2_b, scb,
         mout + (size_t)b * 64 * HW, 64, HW, 1, 0);
    cx.off = mk2;
  }

  // ---- ADWM fusion: out = (cur + mout) + fuse((1-a)*cur + a*mout) ----
  float* p1 = cx.alloc(NT);
  float* p2 = cx.alloc(NT);
  gemm(cx, cur, 64, 0, 0, 0, 0, aw_cx_w, aw_cx_b, nullptr, p1, 64, HW, B_, 0);
  gemm(cx, mout, 64, 0, 0, 0, 0, aw_cy_w, aw_cy_b, nullptr, p2, 64, HW, B_, 0);
  k_mul<<<cdivll(NT, TPB), TPB, 0, stream>>>(p1, p2, p1, NT);
  float* a = cx.alloc((long long)B_ * HW);
  gemm(cx, p1, 64, 0, 0, 0, 0, aw_ca_w, aw_ca_b, nullptr, a, 1, HW, B_, 2);  // sigmoid
  float* idt = cx.alloc(NT);
  k_add<<<cdivll(NT, TPB), TPB, 0, stream>>>(cur, mout, idt, NT);
  float* z = p2;  // reuse
  k_mix<<<cdivll(NT, TPB), TPB, 0, stream>>>(cur, mout, a, z, 64, HW, NT);
  gemm(cx, z, 64, 0, 0, 0, 0, aw_fu_w, aw_fu_b, idt, (float*)d_out, 64, HW, B_, 0);
  cx.off = mk;
}